// AttentionLayer_14955076125481
// MI455X (gfx1250) — compile-verified
//
#include <hip/hip_runtime.h>
#include <hip/hip_bf16.h>
#include <math.h>

// Problem constants (match reference)
#define B 32
#define S 4096
#define D 1024
#define SPLITS 16            // S-splits for the flash pass: 32*16 = 512 workgroups
#define WAVES_PER_WG 8       // 256 threads = 8 wave32s

typedef float v2f __attribute__((ext_vector_type(2)));
typedef float v4f __attribute__((ext_vector_type(4)));
typedef float v8f __attribute__((ext_vector_type(8)));

// ---------------------------------------------------------------------------
// Kernel 1: dec_proj = dec @ W_a  via V_WMMA_F32_16X16X4_F32
// One wave32 per 16x16 output tile. grid = (M/16, N/16) = (2, 64).
// A (16x4 f32) lane layout:  lanes 0-15: M=lane,   K={0,1} in vgpr{0,1}
//                            lanes16-31: M=lane-16,K={2,3} in vgpr{0,1}
// B (4x16 f32) lane layout:  vgpr0: K=0 (lanes0-15) / K=2 (lanes16-31)
//                            vgpr1: K=1 (lanes0-15) / K=3 (lanes16-31)
// C/D (16x16 f32): vgpr v -> row M = v + 8*(lane/16), col N = lane%16
// ---------------------------------------------------------------------------
__global__ __launch_bounds__(32) void proj_wmma(const float* __restrict__ dec,
                                                const float* __restrict__ W,
                                                float* __restrict__ dec_proj) {
  const int mt   = blockIdx.x;        // 0..1
  const int nt   = blockIdx.y;        // 0..63
  const int lane = threadIdx.x;       // 0..31
  const int half = lane >> 4;
  const int lrow = lane & 15;

  const int arow_idx = mt * 16 + lrow;     // A-matrix row this lane supplies
  const int col      = nt * 16 + lrow;     // B/C column this lane supplies

  const float* __restrict__ arow = dec + (size_t)arow_idx * D;

  v8f c = {};
#pragma unroll 4
  for (int k0 = 0; k0 < D; k0 += 4) {
    const int k = k0 + 2 * half;
    v2f a;
    a.x = arow[k];
    a.y = arow[k + 1];
    v2f b;
    b.x = W[(size_t)k * D + col];
    b.y = W[(size_t)(k + 1) * D + col];
    // 8-arg form: (neg_a, A, neg_b, B, c_mod, C, reuse_a, reuse_b)
    c = __builtin_amdgcn_wmma_f32_16x16x4_f32(false, a, false, b,
                                              (short)0, c, false, false);
  }

#pragma unroll
  for (int v = 0; v < 8; ++v) {
    dec_proj[(size_t)(mt * 16 + v + 8 * half) * D + col] = c[v];
  }
}

// ---------------------------------------------------------------------------
// Kernel 2: fused online-softmax partial pass (flash style, split over S).
// grid = (B, SPLITS), block = 256 (8 wave32s). Each wave handles 32 contiguous
// sequence rows: dot(dec_proj, enc_row) -> cross-lane reduce -> online softmax
// with per-lane v4f[8] context accumulator. Waves merge through LDS with
// ds_add_f32 atomics. Emits (m, l, acc[D]) per (b, split) into workspace.
// ---------------------------------------------------------------------------
__global__ __launch_bounds__(256) void flash_partial(const float* __restrict__ enc,
                                                     const float* __restrict__ dec_proj,
                                                     float* __restrict__ pacc,
                                                     float* __restrict__ pm,
                                                     float* __restrict__ pl) {
  const int b     = blockIdx.x;
  const int split = blockIdx.y;
  const int tid   = threadIdx.x;
  const int wave  = tid >> 5;
  const int lane  = tid & 31;

  __shared__ float sm[WAVES_PER_WG];
  __shared__ float sl[WAVES_PER_WG];
  __shared__ float sacc[D];

  for (int i = tid; i < D; i += 256) sacc[i] = 0.0f;
  __syncthreads();

  // Cache this lane's 32 dec_proj values (lane covers d = i*128 + lane*4 .. +3)
  const v4f* __restrict__ dpp = (const v4f*)(dec_proj + (size_t)b * D);
  v4f dp[8];
#pragma unroll
  for (int i = 0; i < 8; ++i) dp[i] = dpp[i * 32 + lane];

  float m = -INFINITY;
  float l = 0.0f;
  v4f acc[8];
#pragma unroll
  for (int i = 0; i < 8; ++i) acc[i] = (v4f){0.f, 0.f, 0.f, 0.f};

  const int s_per_split = S / SPLITS;                    // 256
  const int s_per_wave  = s_per_split / WAVES_PER_WG;    // 32
  const int sbase = split * s_per_split + wave * s_per_wave;

  for (int j = 0; j < s_per_wave; ++j) {
    const int s = sbase + j;
    const v4f* __restrict__ ep =
        (const v4f*)(enc + ((size_t)b * S + (size_t)s) * D);

    v4f e[8];
#pragma unroll
    for (int i = 0; i < 8; ++i)
      e[i] = __builtin_nontemporal_load(&ep[i * 32 + lane]);  // NT: 512MB stream >> L2

    float part = 0.0f;
#pragma unroll
    for (int i = 0; i < 8; ++i) {
      part += dp[i].x * e[i].x + dp[i].y * e[i].y +
              dp[i].z * e[i].z + dp[i].w * e[i].w;
    }
    // wave32 cross-lane sum
#pragma unroll
    for (int off = 16; off > 0; off >>= 1)
      part += __shfl_xor(part, off, 32);

    const float score = part;
    const float nm    = fmaxf(m, score);
    const float scale = __expf(m - nm);      // 0 on first iter (m = -inf)
    const float p     = __expf(score - nm);
    l = l * scale + p;
#pragma unroll
    for (int i = 0; i < 8; ++i) {
      acc[i].x = acc[i].x * scale + p * e[i].x;
      acc[i].y = acc[i].y * scale + p * e[i].y;
      acc[i].z = acc[i].z * scale + p * e[i].z;
      acc[i].w = acc[i].w * scale + p * e[i].w;
    }
    m = nm;
  }

  if (lane == 0) { sm[wave] = m; sl[wave] = l; }
  __syncthreads();

  float M = sm[0];
#pragma unroll
  for (int w = 1; w < WAVES_PER_WG; ++w) M = fmaxf(M, sm[w]);
  float L = 0.0f;
#pragma unroll
  for (int w = 0; w < WAVES_PER_WG; ++w) L += sl[w] * __expf(sm[w] - M);

  const float f = __expf(m - M);   // this wave's rescale factor
#pragma unroll
  for (int i = 0; i < 8; ++i) {
    const int d0 = i * 128 + lane * 4;
    atomicAdd(&sacc[d0 + 0], f * acc[i].x);   // ds_add_f32
    atomicAdd(&sacc[d0 + 1], f * acc[i].y);
    atomicAdd(&sacc[d0 + 2], f * acc[i].z);
    atomicAdd(&sacc[d0 + 3], f * acc[i].w);
  }
  __syncthreads();

  v4f* __restrict__ outp =
      (v4f*)(pacc + (size_t)(b * SPLITS + split) * D);
  const v4f* __restrict__ sa = (const v4f*)sacc;
  for (int i = tid; i < D / 4; i += 256) outp[i] = sa[i];
  if (tid == 0) {
    pm[b * SPLITS + split] = M;
    pl[b * SPLITS + split] = L;
  }
}

// ---------------------------------------------------------------------------
// Kernel 3: merge the SPLITS partials per batch (log-sum-exp rescale) and
// write context[b, :]. grid = B, block = 256 (each thread owns 4 d's).
// ---------------------------------------------------------------------------
__global__ __launch_bounds__(256) void combine(const float* __restrict__ pacc,
                                               const float* __restrict__ pm,
                                               const float* __restrict__ pl,
                                               float* __restrict__ out) {
  const int b   = blockIdx.x;
  const int tid = threadIdx.x;

  float M = -INFINITY;
#pragma unroll
  for (int s = 0; s < SPLITS; ++s) M = fmaxf(M, pm[b * SPLITS + s]);
  float L = 0.0f;
#pragma unroll
  for (int s = 0; s < SPLITS; ++s)
    L += pl[b * SPLITS + s] * __expf(pm[b * SPLITS + s] - M);
  const float inv = 1.0f / L;

  v4f accv = (v4f){0.f, 0.f, 0.f, 0.f};
#pragma unroll
  for (int s = 0; s < SPLITS; ++s) {
    const float f = __expf(pm[b * SPLITS + s] - M) * inv;
    const v4f* __restrict__ pa =
        (const v4f*)(pacc + (size_t)(b * SPLITS + s) * D);
    const v4f v = pa[tid];
    accv.x += f * v.x;
    accv.y += f * v.y;
    accv.z += f * v.z;
    accv.w += f * v.w;
  }
  ((v4f*)(out + (size_t)b * D))[tid] = accv;
}

// ---------------------------------------------------------------------------
// Launch: inputs are decoder_output [B,D], encoder_output [B,S,D], W_a [D,D].
// Workspace layout (floats): dec_proj[B*D] | pacc[B*SPLITS*D] | pm | pl
// Total ~2.25 MB.
// ---------------------------------------------------------------------------
extern "C" void kernel_launch(void* const* d_in, const int* in_sizes, int n_in,
                              void* d_out, int out_size, void* d_ws, size_t ws_size,
                              hipStream_t stream) {
  const float* dec = (const float*)d_in[0];
  const float* enc = (const float*)d_in[1];
  const float* W   = (const float*)d_in[2];
  float* out = (float*)d_out;

  float* wsf      = (float*)d_ws;
  float* dec_proj = wsf;                            // B*D            = 32768
  float* pacc     = dec_proj + (size_t)B * D;       // B*SPLITS*D     = 524288
  float* pm       = pacc + (size_t)B * SPLITS * D;  // B*SPLITS       = 512
  float* pl       = pm + B * SPLITS;                // B*SPLITS       = 512

  proj_wmma<<<dim3(B / 16, D / 16), 32, 0, stream>>>(dec, W, dec_proj);
  flash_partial<<<dim3(B, SPLITS), 256, 0, stream>>>(enc, dec_proj, pacc, pm, pl);
  combine<<<B, 256, 0, stream>>>(pacc, pm, pl, out);
}